// GraphSAGE_89936615178305
// MI455X (gfx1250) — compile-verified
//
#include <hip/hip_runtime.h>
#include <hip/hip_bf16.h>

typedef __attribute__((ext_vector_type(2))) float v2f;
typedef __attribute__((ext_vector_type(8))) float v8f;

#define D_FEAT 64
#define W_PITCH 66           // LDS row pitch to avoid 16-way bank conflicts
#define WOFF (64 * W_PITCH)  // offset of second weight matrix in LDS

// ---------------------------------------------------------------------------
// Zero a float buffer (grid-stride)
// ---------------------------------------------------------------------------
__global__ void zero_f32(float* __restrict__ p, int n) {
    int i = blockIdx.x * blockDim.x + threadIdx.x;
    int stride = gridDim.x * blockDim.x;
    for (; i < n; i += stride) p[i] = 0.0f;
}

// ---------------------------------------------------------------------------
// Edge scatter: one wave (32 lanes) per edge; lane l handles features l, l+32.
// Gather from feat[src] (coalesced 2x128B), atomic-add into msg[dst].
// Optionally accumulates float degree (lane 0 only).
// ---------------------------------------------------------------------------
__global__ void sage_scatter(const float* __restrict__ feat,
                             const long long* __restrict__ ei,
                             float* __restrict__ msg,
                             float* __restrict__ deg,
                             int nEdges, int addDeg) {
    int t = blockIdx.x * blockDim.x + threadIdx.x;
    int e = t >> 5;
    int lane = t & 31;
    if (e >= nEdges) return;
    int src = (int)ei[e];
    int dst = (int)ei[(size_t)nEdges + e];
    float v0 = feat[(size_t)src * D_FEAT + lane];
    float v1 = feat[(size_t)src * D_FEAT + 32 + lane];
    atomicAdd(&msg[(size_t)dst * D_FEAT + lane], v0);
    atomicAdd(&msg[(size_t)dst * D_FEAT + 32 + lane], v1);
    if (addDeg && lane == 0) atomicAdd(&deg[dst], 1.0f);
}

// ---------------------------------------------------------------------------
// In-place: deg[n] = 1 / max(deg[n], 1)
// ---------------------------------------------------------------------------
__global__ void inv_deg(float* __restrict__ deg, int n) {
    int i = blockIdx.x * blockDim.x + threadIdx.x;
    if (i < n) {
        float d = deg[i];
        deg[i] = 1.0f / (d > 1.0f ? d : 1.0f);
    }
}

// ---------------------------------------------------------------------------
// Fused SAGEConv GEMM:  out = (msg * invdeg) @ Wl^T + bias + xin @ Wr^T
// Optionally ReLU. Uses V_WMMA_F32_16X16X4_F32, fused K = 128 (64 agg + 64 root).
// Block = 256 threads = 8 waves; each wave owns a 16-row x 64-col output strip
// (four 16x16 accumulators). Block covers 128 rows.
// ---------------------------------------------------------------------------
__global__ void sage_gemm_wmma(const float* __restrict__ msg,
                               const float* __restrict__ xin,
                               const float* __restrict__ invdeg,
                               const float* __restrict__ Wl,
                               const float* __restrict__ Wr,
                               const float* __restrict__ bias,
                               float* __restrict__ out,
                               int nRows, int doRelu) {
    __shared__ float lw[2 * 64 * W_PITCH];

    int tid = threadIdx.x;
    // Stage both weight matrices into LDS with padded pitch (bank-conflict free
    // B-fragment reads: addr/4 mod 64 varies with output column).
    for (int i = tid; i < 64 * 64; i += 256) {
        int j = i >> 6;      // output feature (row of W)
        int k = i & 63;      // input feature (col of W)
        lw[j * W_PITCH + k] = Wl[i];
        lw[WOFF + j * W_PITCH + k] = Wr[i];
    }
    __syncthreads();

    int wave = tid >> 5;
    int lane = tid & 31;
    int l16 = lane & 15;
    int hi = lane >> 4;                       // 0: K pair {0,1}, 1: K pair {2,3}
    int rowBase = blockIdx.x * 128 + wave * 16;
    int rowA = rowBase + l16;
    int rowC = rowA < nRows ? rowA : (nRows - 1);   // clamp loads, no divergence

    float invd = invdeg[rowC];
    const float* aBase = msg + (size_t)rowC * D_FEAT;
    const float* xBase = xin + (size_t)rowC * D_FEAT;

    v8f acc[4];
    acc[0] = (v8f){}; acc[1] = (v8f){}; acc[2] = (v8f){}; acc[3] = (v8f){};

    // ---- K = 0..63 : aggregated neighbors through Wl ----
#pragma unroll
    for (int kk = 0; kk < 16; ++kk) {
        int klo = kk * 4 + hi * 2;            // even -> 8B-aligned pair
        v2f a;
        a.x = aBase[klo] * invd;
        a.y = aBase[klo + 1] * invd;
#pragma unroll
        for (int ct = 0; ct < 4; ++ct) {
            int col = ct * 16 + l16;
            v2f b;
            b.x = lw[col * W_PITCH + klo];
            b.y = lw[col * W_PITCH + klo + 1];
            acc[ct] = __builtin_amdgcn_wmma_f32_16x16x4_f32(
                false, a, false, b, (short)0, acc[ct], false, false);
        }
    }

    // ---- K = 64..127 : root features through Wr ----
#pragma unroll
    for (int kk = 0; kk < 16; ++kk) {
        int klo = kk * 4 + hi * 2;
        v2f a;
        a.x = xBase[klo];
        a.y = xBase[klo + 1];
#pragma unroll
        for (int ct = 0; ct < 4; ++ct) {
            int col = ct * 16 + l16;
            v2f b;
            b.x = lw[WOFF + col * W_PITCH + klo];
            b.y = lw[WOFF + col * W_PITCH + klo + 1];
            acc[ct] = __builtin_amdgcn_wmma_f32_16x16x4_f32(
                false, a, false, b, (short)0, acc[ct], false, false);
        }
    }

    // ---- bias + (ReLU) + predicated store ----
    // C/D layout: VGPR r, lanes 0-15 -> (M=r, N=lane), lanes 16-31 -> (M=r+8)
#pragma unroll
    for (int ct = 0; ct < 4; ++ct) {
        int col = ct * 16 + l16;
        float bv = bias[col];
#pragma unroll
        for (int r = 0; r < 8; ++r) {
            int rowO = rowBase + r + hi * 8;
            float v = acc[ct][r] + bv;
            if (doRelu) v = v > 0.0f ? v : 0.0f;
            if (rowO < nRows) out[(size_t)rowO * D_FEAT + col] = v;
        }
    }
}

// ---------------------------------------------------------------------------
extern "C" void kernel_launch(void* const* d_in, const int* in_sizes, int n_in,
                              void* d_out, int out_size, void* d_ws, size_t ws_size,
                              hipStream_t stream) {
    const float* x        = (const float*)d_in[0];
    const long long* ei   = (const long long*)d_in[1];
    const float* W1l      = (const float*)d_in[2];
    const float* b1l      = (const float*)d_in[3];
    const float* W1r      = (const float*)d_in[4];
    const float* W2l      = (const float*)d_in[5];
    const float* b2l      = (const float*)d_in[6];
    const float* W2r      = (const float*)d_in[7];

    const int nNodes = in_sizes[0] / D_FEAT;   // 100000
    const int nEdges = in_sizes[1] / 2;        // 1000000
    const int featN  = nNodes * D_FEAT;        // 6.4M floats

    float* msg = (float*)d_ws;                 // [nNodes, 64]
    float* h   = msg + featN;                  // [nNodes, 64]
    float* deg = h + featN;                    // [nNodes] -> becomes 1/deg

    float* outp = (float*)d_out;

    const int zGrid   = (featN + 255) / 256;
    const int dGrid   = (nNodes + 255) / 256;
    const int sGrid   = (nEdges * 32 + 255) / 256;
    const int gGrid   = (nNodes + 127) / 128;

    // Layer 1
    zero_f32<<<zGrid, 256, 0, stream>>>(msg, featN);
    zero_f32<<<dGrid, 256, 0, stream>>>(deg, nNodes);
    sage_scatter<<<sGrid, 256, 0, stream>>>(x, ei, msg, deg, nEdges, 1);
    inv_deg<<<dGrid, 256, 0, stream>>>(deg, nNodes);
    sage_gemm_wmma<<<gGrid, 256, 0, stream>>>(msg, x, deg, W1l, W1r, b1l,
                                              h, nNodes, 1);
    // Layer 2 (reuse msg buffer)
    zero_f32<<<zGrid, 256, 0, stream>>>(msg, featN);
    sage_scatter<<<sGrid, 256, 0, stream>>>(h, ei, msg, deg, nEdges, 0);
    sage_gemm_wmma<<<gGrid, 256, 0, stream>>>(msg, h, deg, W2l, W2r, b2l,
                                              outp, nNodes, 0);
}